// GATWrapper_86870008529628
// MI455X (gfx1250) — compile-verified
//
#include <hip/hip_runtime.h>
#include <hip/hip_bf16.h>
#include <math.h>

typedef __attribute__((ext_vector_type(2))) float v2f;
typedef __attribute__((ext_vector_type(8))) float v8f;
typedef __attribute__((ext_vector_type(4))) unsigned u32x4;
typedef __attribute__((ext_vector_type(4))) int i32x4;
typedef __attribute__((ext_vector_type(8))) int i32x8;

#define N_NODES 100000
#define N_EDGES 1600000
#define IN_DIM 128
#define HEADS 4
#define OUT_DIM 32
#define OUTC 128 /* HEADS*OUT_DIM */
#define NEG_SLOPE 0.2f

// ---- order-preserving float <-> uint key for atomicMax-based segment max ----
__device__ __forceinline__ unsigned fkey(float f) {
  unsigned u = __float_as_uint(f);
  return (u & 0x80000000u) ? ~u : (u | 0x80000000u);
}
__device__ __forceinline__ float funkey(unsigned k) {
  unsigned u = (k & 0x80000000u) ? (k & 0x7fffffffu) : ~k;
  return __uint_as_float(u);
}

// ---------------- GEMM: h = x @ W via V_WMMA_F32_16X16X4_F32 ----------------
// W (128x128 fp32, 64KB) staged in LDS via the Tensor Data Mover (TDM);
// per-wave A tile (16 rows x 128 K) preloaded into 64 VGPRs while the TDM
// transfer is in flight, then reused across all 8 N-subtiles.
__global__ __launch_bounds__(256) void gat_gemm_wmma(const float* __restrict__ x,
                                                     const float* __restrict__ W,
                                                     float* __restrict__ h) {
  __shared__ float lW[IN_DIM * IN_DIM]; // 64KB fp32, linear row-major (TDM fill)
  const int t = threadIdx.x;

  // Phantom store the compiler cannot prove dead: guarantees "a store to lW
  // exists in the module", defeating the never-stored => undef load fold.
  if (blockIdx.x == 0x7fffffffu) lW[t] = 0.f;

#if __has_builtin(__builtin_amdgcn_tensor_load_to_lds)
  if (t < 32) {  // one TDM op, issued by wave 0 only (TDM ignores EXEC)
    const unsigned long long ga = (unsigned long long)(uintptr_t)W;
    u32x4 g0;
    g0[0] = 1u;                                   // count=1, user descriptor
    g0[1] = 0u;                                   // lds_addr: lW at LDS base
    g0[2] = (unsigned)(ga & 0xffffffffu);         // global_addr[31:0]
    g0[3] = (unsigned)((ga >> 32) & 0x01ffffffu)  // global_addr[56:32]
            | (2u << 30);                         // type=2 ("image")
    i32x8 g1;
    g1[0] = (int)(2u << 16);    // wg_mask=0, data_size=2 (4 bytes)
    g1[1] = (int)(128u << 16);  // tensor_dim0=128 (low16 @ bit48)
    g1[2] = (int)(128u << 16);  // tensor_dim0 hi=0 | tensor_dim1=128 (low16)
    g1[3] = (int)(128u << 16);  // tensor_dim1 hi=0 | tile_dim0=128
    g1[4] = (int)128;           // tile_dim1=128, tile_dim2=0 (2D)
    g1[5] = (int)128;           // tensor_dim0_stride = 128 (low32)
    g1[6] = 0;                  // stride0 hi16 | tensor_dim1_stride low16
    g1[7] = 0;                  // tensor_dim1_stride hi32 (unused, 2D)
    const i32x4 z4 = {0, 0, 0, 0};
    const i32x8 z8 = {0, 0, 0, 0, 0, 0, 0, 0};
    // 6-arg form (clang-23 / therock-10.0 headers)
    __builtin_amdgcn_tensor_load_to_lds(g0, g1, z4, z4, z8, 0);
  }
#else
  for (int c4 = t; c4 < (IN_DIM * IN_DIM) / 4; c4 += 256) {
    ((float4*)lW)[c4] = ((const float4*)W)[c4];
  }
#endif

  const int wave = t >> 5;
  const int lane = t & 31;
  const int half = lane >> 4;  // selects K pair per 16x4 fp32 A/B layout
  const int l = lane & 15;
  const int rowBase = blockIdx.x * 128 + wave * 16;
  const int row = rowBase + l;  // A-matrix M index for this lane
  // Clamp instead of branching: an A row only feeds its own C row (same M),
  // and OOB C rows are dropped at the store. Keeps EXEC uniform, no saveexec.
  const int rrow = row < N_NODES ? row : N_NODES - 1;
  const float* ap = x + (size_t)rrow * IN_DIM + 2 * half;

  // Preload whole A tile while the TDM fill is in flight:
  // 32 K-steps x float2 per lane, reused for all 8 nt subtiles.
  v2f a[32];
  #pragma unroll
  for (int kk = 0; kk < 32; ++kk) {
    const float2 a2 = *(const float2*)(ap + kk * 4);
    a[kk][0] = a2.x;
    a[kk][1] = a2.y;
  }

#if __has_builtin(__builtin_amdgcn_tensor_load_to_lds)
  if (t < 32) {
    __builtin_amdgcn_s_wait_tensorcnt(0);
  }
#endif
  // The TDM writes LDS through a descriptor, invisibly to alias analysis.
  // Escape lW's address into opaque asm with a memory clobber so the
  // compiler must assume it has been written (keeps the ds_loads live).
  asm volatile("" : : "r"(lW) : "memory");
  __syncthreads();

  for (int nt = 0; nt < 8; ++nt) {
    const int n0 = nt * 16;
    v8f c = {};
    #pragma unroll
    for (int kk = 0; kk < 32; ++kk) {
      v2f B;
      B[0] = lW[(kk * 4 + 2 * half) * IN_DIM + n0 + l];
      B[1] = lW[(kk * 4 + 2 * half + 1) * IN_DIM + n0 + l];
      // 8 args: (neg_a, A, neg_b, B, c_mod, C, reuse_a, reuse_b)
      c = __builtin_amdgcn_wmma_f32_16x16x4_f32(false, a[kk], false, B,
                                                (short)0, c, false, false);
    }
    // C/D layout: M = v + 8*half, N = l
    #pragma unroll
    for (int v = 0; v < 8; ++v) {
      const int r = rowBase + v + 8 * half;
      if (r < N_NODES) h[(size_t)r * OUTC + n0 + l] = c[v];
    }
  }
}

// ---------------- per-node attention dot products ----------------
__global__ void gat_attn_dots(const float* __restrict__ h,
                              const float* __restrict__ att_src,
                              const float* __restrict__ att_dst,
                              float* __restrict__ a_src,
                              float* __restrict__ a_dst) {
  const int idx = blockIdx.x * blockDim.x + threadIdx.x;
  if (idx >= N_NODES * HEADS) return;
  const int n = idx >> 2, hh = idx & 3;
  const float4* hp = (const float4*)(h + (size_t)n * OUTC + hh * OUT_DIM);
  const float4* as = (const float4*)(att_src + hh * OUT_DIM);
  const float4* ad = (const float4*)(att_dst + hh * OUT_DIM);
  float s = 0.f, d = 0.f;
  #pragma unroll
  for (int i = 0; i < OUT_DIM / 4; ++i) {
    const float4 hv = hp[i], av = as[i], bv = ad[i];
    s += hv.x * av.x + hv.y * av.y + hv.z * av.z + hv.w * av.w;
    d += hv.x * bv.x + hv.y * bv.y + hv.z * bv.z + hv.w * bv.w;
  }
  a_src[idx] = s;
  a_dst[idx] = d;
}

// ---------------- init: zero output accum, emax keys, denom ----------------
__global__ void gat_init(float* __restrict__ out, unsigned* __restrict__ emaxK,
                         float* __restrict__ den) {
  const int i = blockIdx.x * blockDim.x + threadIdx.x;
  if (i < N_NODES * OUTC) out[i] = 0.f;
  if (i < N_NODES * HEADS) {
    emaxK[i] = 0u;  // below any finite key; every dst has a self-loop
    den[i] = 0.f;
  }
}

__device__ __forceinline__ void edge_pair(const int* __restrict__ ei, int e,
                                          int& src, int& dst) {
  if (e < N_EDGES) {
    src = ei[e];
    dst = ei[N_EDGES + e];
  } else {
    src = dst = e - N_EDGES;  // self-loops appended, PyG GATConv default
  }
}

// ---------------- pass 1: segment max of leaky-relu logits ----------------
__global__ void gat_edge_max(const int* __restrict__ ei,
                             const float* __restrict__ a_src,
                             const float* __restrict__ a_dst,
                             unsigned* __restrict__ emaxK) {
  const int idx = blockIdx.x * blockDim.x + threadIdx.x;
  if (idx >= (N_EDGES + N_NODES) * HEADS) return;
  const int e = idx >> 2, hh = idx & 3;
  int src, dst;
  edge_pair(ei, e, src, dst);
  float v = a_src[src * HEADS + hh] + a_dst[dst * HEADS + hh];
  v = v < 0.f ? NEG_SLOPE * v : v;
  atomicMax(&emaxK[dst * HEADS + hh], fkey(v));
}

// ---------------- pass 2: softmax denominator ----------------
__global__ void gat_edge_den(const int* __restrict__ ei,
                             const float* __restrict__ a_src,
                             const float* __restrict__ a_dst,
                             const unsigned* __restrict__ emaxK,
                             float* __restrict__ den) {
  const int idx = blockIdx.x * blockDim.x + threadIdx.x;
  if (idx >= (N_EDGES + N_NODES) * HEADS) return;
  const int e = idx >> 2, hh = idx & 3;
  int src, dst;
  edge_pair(ei, e, src, dst);
  float v = a_src[src * HEADS + hh] + a_dst[dst * HEADS + hh];
  v = v < 0.f ? NEG_SLOPE * v : v;
  const float m = funkey(emaxK[dst * HEADS + hh]);
  atomicAdd(&den[dst * HEADS + hh], expf(v - m));
}

// ---------------- pass 3: wave-per-edge weighted scatter ----------------
__global__ __launch_bounds__(256) void gat_scatter(
    const int* __restrict__ ei, const float* __restrict__ h,
    const float* __restrict__ a_src, const float* __restrict__ a_dst,
    const unsigned* __restrict__ emaxK, const float* __restrict__ den,
    float* __restrict__ out) {
  const long long gt = (long long)blockIdx.x * blockDim.x + threadIdx.x;
  const int edge = (int)(gt >> 5);
  if (edge >= N_EDGES + N_NODES) return;
  const int lane = threadIdx.x & 31;  // lane == feature dim within a head
  int src, dst;
  edge_pair(ei, edge, src, dst);
  __builtin_prefetch(h + (size_t)src * OUTC, 0, 0);  // global_prefetch_b8
  #pragma unroll
  for (int hh = 0; hh < HEADS; ++hh) {
    float v = a_src[src * HEADS + hh] + a_dst[dst * HEADS + hh];
    v = v < 0.f ? NEG_SLOPE * v : v;
    const float m = funkey(emaxK[dst * HEADS + hh]);
    const float alpha = expf(v - m) / (den[dst * HEADS + hh] + 1e-16f);
    const float msg = h[(size_t)src * OUTC + hh * OUT_DIM + lane] * alpha;
    atomicAdd(&out[(size_t)dst * OUTC + hh * OUT_DIM + lane], msg);
  }
}

// ---------------- finalize: + bias, ELU ----------------
__global__ void gat_finalize(float* __restrict__ out,
                             const float* __restrict__ bias) {
  const int i = blockIdx.x * blockDim.x + threadIdx.x;
  if (i >= N_NODES * OUTC) return;
  const float v = out[i] + bias[i & 127];
  out[i] = v > 0.f ? v : expm1f(v);
}

extern "C" void kernel_launch(void* const* d_in, const int* in_sizes, int n_in,
                              void* d_out, int out_size, void* d_ws,
                              size_t ws_size, hipStream_t stream) {
  const float* x = (const float*)d_in[0];
  const int* ei = (const int*)d_in[1];  // edge_index [2, E]
  const float* W = (const float*)d_in[2];
  const float* att_src = (const float*)d_in[3];
  const float* att_dst = (const float*)d_in[4];
  const float* bias = (const float*)d_in[5];
  float* out = (float*)d_out;

  // workspace layout
  float* h = (float*)d_ws;                               // N*128
  float* a_src = h + (size_t)N_NODES * OUTC;             // N*4
  float* a_dst = a_src + (size_t)N_NODES * HEADS;        // N*4
  unsigned* emaxK = (unsigned*)(a_dst + (size_t)N_NODES * HEADS);  // N*4
  float* den = (float*)(emaxK + (size_t)N_NODES * HEADS);          // N*4

  const int T = 256;
  const int initN = N_NODES * OUTC;
  gat_init<<<(initN + T - 1) / T, T, 0, stream>>>(out, emaxK, den);

  gat_gemm_wmma<<<(N_NODES + 127) / 128, T, 0, stream>>>(x, W, h);

  const int dots = N_NODES * HEADS;
  gat_attn_dots<<<(dots + T - 1) / T, T, 0, stream>>>(h, att_src, att_dst,
                                                      a_src, a_dst);

  const int eh = (N_EDGES + N_NODES) * HEADS;
  gat_edge_max<<<(eh + T - 1) / T, T, 0, stream>>>(ei, a_src, a_dst, emaxK);
  gat_edge_den<<<(eh + T - 1) / T, T, 0, stream>>>(ei, a_src, a_dst, emaxK,
                                                   den);

  const long long sthreads = (long long)(N_EDGES + N_NODES) * 32;
  gat_scatter<<<(unsigned)((sthreads + T - 1) / T), T, 0, stream>>>(
      ei, h, a_src, a_dst, emaxK, den, out);

  gat_finalize<<<(initN + T - 1) / T, T, 0, stream>>>(out, bias);
}